// NodeFeatures_549755814507
// MI455X (gfx1250) — compile-verified
//
#include <hip/hip_runtime.h>

typedef __attribute__((ext_vector_type(2))) float v2f;
typedef __attribute__((ext_vector_type(4))) float v4f;
typedef __attribute__((ext_vector_type(8))) float v8f;

#define B_ 8
#define N_ 128
#define H_ 512
#define ROWS_ (B_ * N_)          // 1024

// ---------------------------------------------------------------------------
// Kernel 1: Ux = x @ Uw^T + Ub ; Vx = x @ Vw^T + Vb
// One wave32 per 16x16 output tile, V_WMMA_F32_16X16X4_F32, K-loop over H.
//   D[m,n] = sum_k A[m,k] * B[k,n],  A[m,k] = x[row0+m, h+k],
//   B[k,n] = W[col0+n, h+k]   (reference computes x @ W.T)
// f32 16x4 A-fragment layout: lanes 0-15 -> M=lane, v0=K0,v1=K1;
//                             lanes16-31 -> M=lane-16, v0=K2,v1=K3.
// B mirrors with N in place of M. Both fragments = one contiguous b64 load.
// ---------------------------------------------------------------------------
__global__ __launch_bounds__(128) void gemm_wmma_f32(
    const float* __restrict__ x,
    const float* __restrict__ Uw, const float* __restrict__ Ub,
    const float* __restrict__ Vw, const float* __restrict__ Vb,
    float* __restrict__ Ux, float* __restrict__ Vx)
{
    const int lane = threadIdx.x & 31;
    const int wave = threadIdx.x >> 5;

    int tile = blockIdx.x * 4 + wave;        // 0..4095  (2 mats * 64 * 32)
    const int mat     = tile >> 11;          // 0 = U, 1 = V
    tile &= 2047;
    const int rowTile = tile >> 5;           // 0..63 over B*N rows
    const int colTile = tile & 31;           // 0..31 over H cols

    const float* __restrict__ W    = mat ? Vw : Uw;
    const float* __restrict__ bias = mat ? Vb : Ub;
    float*       __restrict__ Out  = mat ? Vx : Ux;

    const int l15   = lane & 15;
    const int kbase = (lane < 16) ? 0 : 2;
    const int row   = rowTile * 16 + l15;    // A row for this lane
    const int col   = colTile * 16 + l15;    // B col / bias / D col

    const float* __restrict__ arow = x + (size_t)row * H_ + kbase;
    const float* __restrict__ brow = W + (size_t)col * H_ + kbase;

    v8f c = {};
#pragma unroll 8
    for (int h = 0; h < H_; h += 4) {
        v2f a = *(const v2f*)(arow + h);     // K = h+kbase, h+kbase+1
        v2f b = *(const v2f*)(brow + h);
        c = __builtin_amdgcn_wmma_f32_16x16x4_f32(
                /*neg_a=*/false, a, /*neg_b=*/false, b,
                /*c_mod=*/(short)0, c, /*reuse_a=*/false, /*reuse_b=*/false);
    }

    const float bv = bias[col];
    const int mbase = (lane < 16) ? 0 : 8;
#pragma unroll
    for (int r = 0; r < 8; ++r) {
        const int m = mbase + r;
        Out[(size_t)(rowTile * 16 + m) * H_ + col] = c[r] + bv;
    }
}

// ---------------------------------------------------------------------------
// Kernel 2: single streaming pass over edge_gate (256 MB, read-once -> NT
// loads so Vx/Ux stay hot in L2).  One block per (b,i); 128 threads x float4
// over H.  num and den accumulated together; fused epilogue.
// ---------------------------------------------------------------------------
__global__ __launch_bounds__(128) void gate_reduce(
    const float* __restrict__ eg,
    const float* __restrict__ Ux,
    const float* __restrict__ Vx,
    float* __restrict__ out)
{
    const int bi = blockIdx.x;               // b*N + i, 0..1023
    const int t  = threadIdx.x;              // 0..127 -> 4 consecutive h
    const int b  = bi >> 7;                  // / N_

    const v4f* __restrict__ eg4 = (const v4f*)eg + (size_t)bi * (size_t)(N_ * H_ / 4) + t;
    const v4f* __restrict__ vx4 = (const v4f*)Vx + (size_t)b  * (size_t)(N_ * H_ / 4) + t;

    v4f num = {0.f, 0.f, 0.f, 0.f};
    v4f den = {0.f, 0.f, 0.f, 0.f};

#pragma unroll 4
    for (int j = 0; j < N_; ++j) {
        v4f e = __builtin_nontemporal_load(eg4 + (size_t)j * (H_ / 4));
        v4f v = vx4[(size_t)j * (H_ / 4)];
        num += e * v;
        den += e;
    }

    const v4f u = ((const v4f*)Ux)[(size_t)bi * (H_ / 4) + t];
    v4f o;
    o.x = u.x + num.x / (1e-20f + den.x);
    o.y = u.y + num.y / (1e-20f + den.y);
    o.z = u.z + num.z / (1e-20f + den.z);
    o.w = u.w + num.w / (1e-20f + den.w);
    ((v4f*)out)[(size_t)bi * (H_ / 4) + t] = o;
}

// ---------------------------------------------------------------------------
extern "C" void kernel_launch(void* const* d_in, const int* in_sizes, int n_in,
                              void* d_out, int out_size, void* d_ws, size_t ws_size,
                              hipStream_t stream)
{
    const float* x  = (const float*)d_in[0];
    const float* eg = (const float*)d_in[1];
    const float* Uw = (const float*)d_in[2];
    const float* Ub = (const float*)d_in[3];
    const float* Vw = (const float*)d_in[4];
    const float* Vb = (const float*)d_in[5];
    float* out = (float*)d_out;

    float* Ux = (float*)d_ws;                         // 2 MB
    float* Vx = Ux + (size_t)ROWS_ * H_;              // 2 MB

    // 4096 tiles (2 matrices x 64 row-tiles x 32 col-tiles), 4 waves/block
    gemm_wmma_f32<<<1024, 128, 0, stream>>>(x, Uw, Ub, Vw, Vb, Ux, Vx);
    // one block per (b,i) row of edge_gate
    gate_reduce<<<ROWS_, 128, 0, stream>>>(eg, Ux, Vx, out);
}